// Net_45810121179577
// MI455X (gfx1250) — compile-verified
//
#include <hip/hip_runtime.h>
#include <hip/hip_bf16.h>
#include <math.h>

typedef __attribute__((ext_vector_type(2))) float v2f;
typedef __attribute__((ext_vector_type(8))) float v8f;

#define BLK 256

// ---------------- fill ----------------
__global__ void fill_f32(float* __restrict__ p, float v, int n) {
    int i = blockIdx.x * blockDim.x + threadIdx.x;
    if (i < n) p[i] = v;
}

// ---------------- degree: deg[row] += ew (masked self-loops) ----------------
__global__ void deg_kernel(const int* __restrict__ row, const int* __restrict__ col,
                           const float* __restrict__ ew, float* __restrict__ deg, int E) {
    int e = blockIdx.x * blockDim.x + threadIdx.x;
    if (e >= E) return;
    int r = row[e], c = col[e];
    float w = (r == c) ? 0.0f : ew[e];
    unsafeAtomicAdd(&deg[r], w);
}

// ---------------- dis = deg>0 ? rsqrt(deg) : 0   (in place) ----------------
__global__ void dis_kernel(float* __restrict__ deg, int n) {
    int i = blockIdx.x * blockDim.x + threadIdx.x;
    if (i >= n) return;
    float d = deg[i];
    deg[i] = (d > 0.0f) ? rsqrtf(d) : 0.0f;
}

// ---------------- norm[e] = -dis[row]*ew*dis[col] ----------------
__global__ void norm_kernel(const int* __restrict__ row, const int* __restrict__ col,
                            const float* __restrict__ ew, const float* __restrict__ dis,
                            float* __restrict__ norm, int E) {
    int e = blockIdx.x * blockDim.x + threadIdx.x;
    if (e >= E) return;
    int r = row[e], c = col[e];
    float w = (r == c) ? 0.0f : ew[e];
    norm[e] = -dis[r] * w * dis[c];
}

// ---------------- scatter: P[col] += norm * h[row]  (2-wide, b64 gather) ----------------
__global__ void scatter_prop(const int* __restrict__ row, const int* __restrict__ col,
                             const float* __restrict__ norm, const float2* __restrict__ h,
                             float* __restrict__ P, int E) {
    int e = blockIdx.x * blockDim.x + threadIdx.x;
    if (e >= E) return;
    int r = row[e], c = col[e];
    float w = norm[e];
    float2 hv = h[r];                       // single global_load_b64 from L2-resident buffer
    unsafeAtomicAdd(&P[(size_t)c * 2 + 0], w * hv.x);
    unsafeAtomicAdd(&P[(size_t)c * 2 + 1], w * hv.y);
}

// ---------------- Chebyshev node update + WMMA transform ----------------
// mode 0: Tnew = x[:, :2]           ; outacc  = Tnew @ Wk   (init write)
// mode 1: Tnew = P                  ; outacc += Tnew @ Wk
// mode 2: Tnew = 2*P - Tprev        ; outacc += Tnew @ Wk
// Each wave handles 16 nodes via V_WMMA_F32_16X16X4_F32:
//   A (16x4): lanes 0-15 carry {K=0,K=1}; lanes 16-31 carry K=2,3 = 0
//   B (4x16): W_cheb[k] (2x2) padded; only K<2, N<2 nonzero
//   D (16x16): columns N<2 hold the 2 output features; extracted via LDS
__global__ void cheb_node_wmma(const float* __restrict__ src,
                               const float2* __restrict__ Tprev,
                               float2* __restrict__ Tnew,
                               float2* __restrict__ outacc,
                               const float* __restrict__ Wk,
                               int n, int mode) {
    __shared__ float lds[8 * 32];               // 8 waves * 16 nodes * 2 feats
    const int tid  = threadIdx.x;
    const int wave = tid >> 5;
    const int lane = tid & 31;
    const int m16  = lane & 15;
    const int base = (blockIdx.x * 8 + wave) * 16;
    const int node = base + m16;
    const bool valid = node < n;

    float t0 = 0.0f, t1 = 0.0f;
    if (lane < 16 && valid) {
        if (mode == 0) {
            t0 = src[(size_t)node * 3 + 0];
            t1 = src[(size_t)node * 3 + 1];
        } else if (mode == 1) {
            float2 p = ((const float2*)src)[node];
            t0 = p.x; t1 = p.y;
        } else {
            float2 p  = ((const float2*)src)[node];
            float2 tp = Tprev[node];
            t0 = 2.0f * p.x - tp.x;
            t1 = 2.0f * p.y - tp.y;
        }
    }

    // A matrix: VGPR0 = K0 (lanes<16) / K2 (lanes>=16, zero); VGPR1 = K1 / K3
    v2f a;
    a.x = t0;
    a.y = t1;

    // B matrix (4x16): b.x = row K=0 (lanes<16) / K=2 (zero); b.y = K=1 / K=3
    float w00 = Wk[0], w01 = Wk[1], w10 = Wk[2], w11 = Wk[3];
    v2f b;
    bool lo = lane < 16;
    b.x = (lo && m16 == 0) ? w00 : ((lo && m16 == 1) ? w01 : 0.0f);
    b.y = (lo && m16 == 0) ? w10 : ((lo && m16 == 1) ? w11 : 0.0f);

    v8f c = {};
    c = __builtin_amdgcn_wmma_f32_16x16x4_f32(
            /*neg_a=*/false, a, /*neg_b=*/false, b,
            /*c_mod=*/(short)0, c, /*reuse_a=*/false, /*reuse_b=*/false);

    // Extract D[M][N] for N<2: lanes 0,1 hold M=c, lanes 16,17 hold M=c+8
    const int Ncol  = m16;
    const int Mbase = (lane >= 16) ? 8 : 0;
    if (Ncol < 2) {
#pragma unroll
        for (int cc = 0; cc < 8; ++cc)
            lds[wave * 32 + (Mbase + cc) * 2 + Ncol] = c[cc];
    }
    __syncthreads();

    if (lane < 16 && valid) {
        float o0 = lds[wave * 32 + m16 * 2 + 0];
        float o1 = lds[wave * 32 + m16 * 2 + 1];
        float2 o;
        if (mode == 0) {
            o.x = o0; o.y = o1;
        } else {
            float2 prev = outacc[node];
            o.x = prev.x + o0; o.y = prev.y + o1;
        }
        outacc[node] = o;                          // global_store_b64
        float2 t; t.x = t0; t.y = t1;
        Tnew[node] = t;                            // global_store_b64
    }
}

// ---------------- u = x_turn*W0 + relu(out+cb)·{W1,W2} + b ----------------
__global__ void finalize_u(const float* __restrict__ x, const float2* __restrict__ outacc,
                           const float* __restrict__ cb, const float* __restrict__ W,
                           const float* __restrict__ b, float* __restrict__ u, int n) {
    int i = blockIdx.x * blockDim.x + threadIdx.x;
    if (i >= n) return;
    float2 o = outacc[i];
    float h0 = fmaxf(o.x + cb[0], 0.0f);
    float h1 = fmaxf(o.y + cb[1], 0.0f);
    u[i] = x[(size_t)i * 3 + 2] * W[0] + h0 * W[1] + h1 * W[2] + b[0];
}

// ---------------- softmax reductions ----------------
__global__ void reduce_max_s1(const float* __restrict__ u, float* __restrict__ part, int n) {
    __shared__ float s[BLK];
    float m = -INFINITY;
    for (int i = blockIdx.x * BLK + threadIdx.x; i < n; i += BLK * gridDim.x)
        m = fmaxf(m, u[i]);
    s[threadIdx.x] = m;
    __syncthreads();
    for (int off = BLK / 2; off > 0; off >>= 1) {
        if (threadIdx.x < off) s[threadIdx.x] = fmaxf(s[threadIdx.x], s[threadIdx.x + off]);
        __syncthreads();
    }
    if (threadIdx.x == 0) part[blockIdx.x] = s[0];
}

__global__ void reduce_max_s2(const float* __restrict__ part, float* __restrict__ mout) {
    __shared__ float s[BLK];
    s[threadIdx.x] = part[threadIdx.x];
    __syncthreads();
    for (int off = BLK / 2; off > 0; off >>= 1) {
        if (threadIdx.x < off) s[threadIdx.x] = fmaxf(s[threadIdx.x], s[threadIdx.x + off]);
        __syncthreads();
    }
    if (threadIdx.x == 0) mout[0] = s[0];
}

__global__ void exp_sum_s1(const float* __restrict__ u, const float* __restrict__ mptr,
                           float* __restrict__ out, float* __restrict__ part, int n) {
    __shared__ float s[BLK];
    float m = mptr[0];
    float acc = 0.0f;
    for (int i = blockIdx.x * BLK + threadIdx.x; i < n; i += BLK * gridDim.x) {
        float e = expf(u[i] - m);
        out[i] = e;
        acc += e;
    }
    s[threadIdx.x] = acc;
    __syncthreads();
    for (int off = BLK / 2; off > 0; off >>= 1) {
        if (threadIdx.x < off) s[threadIdx.x] += s[threadIdx.x + off];
        __syncthreads();
    }
    if (threadIdx.x == 0) part[blockIdx.x] = s[0];
}

__global__ void exp_sum_s2(const float* __restrict__ part, float* __restrict__ invout) {
    __shared__ float s[BLK];
    s[threadIdx.x] = part[threadIdx.x];
    __syncthreads();
    for (int off = BLK / 2; off > 0; off >>= 1) {
        if (threadIdx.x < off) s[threadIdx.x] += s[threadIdx.x + off];
        __syncthreads();
    }
    if (threadIdx.x == 0) invout[0] = 1.0f / s[0];
}

__global__ void scale_kernel(float* __restrict__ out, const float* __restrict__ invptr, int n) {
    int i = blockIdx.x * blockDim.x + threadIdx.x;
    if (i < n) out[i] *= invptr[0];
}

// ---------------- launcher ----------------
extern "C" void kernel_launch(void* const* d_in, const int* in_sizes, int n_in,
                              void* d_out, int out_size, void* d_ws, size_t ws_size,
                              hipStream_t stream) {
    const float* x   = (const float*)d_in[0];   // N*3
    const int*   ei  = (const int*)d_in[1];     // 2*E (row, then col)
    const float* ew  = (const float*)d_in[2];   // E
    const float* Wch = (const float*)d_in[3];   // K*2*2
    const float* cb  = (const float*)d_in[4];   // 2
    const float* W   = (const float*)d_in[5];   // 3
    const float* b   = (const float*)d_in[6];   // 1

    const int N = in_sizes[0] / 3;
    const int E = in_sizes[2];
    const int K = in_sizes[3] / 4;

    const int* row = ei;
    const int* col = ei + E;

    // workspace layout (floats)
    float* w      = (float*)d_ws;
    float* deg    = w;                          // N  (becomes dis in place)
    float* normv  = deg + N;                    // E
    float* Ta     = normv + E;                  // 2N
    float* Tb     = Ta + 2 * (size_t)N;         // 2N
    float* P      = Tb + 2 * (size_t)N;         // 2N
    float* outacc = P + 2 * (size_t)N;          // 2N
    float* u      = outacc + 2 * (size_t)N;     // N
    float* red    = u + N;                      // reduction scratch
    float* maxpart = red;
    float* mval    = red + 256;
    float* sumpart = red + 260;
    float* sinv    = red + 516;

    float* outp = (float*)d_out;

    const int gE  = (E + BLK - 1) / BLK;
    const int gN  = (N + BLK - 1) / BLK;
    const int g2N = (2 * N + BLK - 1) / BLK;
    const int gC  = (N + 127) / 128;   // 8 waves * 16 nodes per block

    // 1) degree + normalization coefficients
    fill_f32<<<gN, BLK, 0, stream>>>(deg, 0.0f, N);
    deg_kernel<<<gE, BLK, 0, stream>>>(row, col, ew, deg, E);
    dis_kernel<<<gN, BLK, 0, stream>>>(deg, N);
    norm_kernel<<<gE, BLK, 0, stream>>>(row, col, ew, deg, normv, E);

    // 2) Chebyshev recursion
    // k=0: Ta = x[:, :2]; outacc = Ta @ W0
    cheb_node_wmma<<<gC, BLK, 0, stream>>>(x, nullptr, (float2*)Ta, (float2*)outacc,
                                           Wch + 0, N, 0);

    // k=1: P = prop(Ta); Tb = P; outacc += Tb @ W1
    fill_f32<<<g2N, BLK, 0, stream>>>(P, 0.0f, 2 * N);
    scatter_prop<<<gE, BLK, 0, stream>>>(row, col, normv, (const float2*)Ta, P, E);
    cheb_node_wmma<<<gC, BLK, 0, stream>>>(P, nullptr, (float2*)Tb, (float2*)outacc,
                                           Wch + 4, N, 1);

    // k=2..K-1: Tnew = 2*prop(Tcur) - Tprev; outacc += Tnew @ Wk
    float* Tprev = Ta;
    float* Tcur  = Tb;
    for (int k = 2; k < K; ++k) {
        fill_f32<<<g2N, BLK, 0, stream>>>(P, 0.0f, 2 * N);
        scatter_prop<<<gE, BLK, 0, stream>>>(row, col, normv, (const float2*)Tcur, P, E);
        cheb_node_wmma<<<gC, BLK, 0, stream>>>(P, (const float2*)Tprev, (float2*)Tprev,
                                               (float2*)outacc, Wch + 4 * k, N, 2);
        float* t = Tprev; Tprev = Tcur; Tcur = t;   // Tcur now holds the new Tx
    }

    // 3) final linear
    finalize_u<<<gN, BLK, 0, stream>>>(x, (const float2*)outacc, cb, W, b, u, N);

    // 4) softmax over all N
    reduce_max_s1<<<256, BLK, 0, stream>>>(u, maxpart, N);
    reduce_max_s2<<<1, BLK, 0, stream>>>(maxpart, mval);
    exp_sum_s1<<<256, BLK, 0, stream>>>(u, mval, outp, sumpart, N);
    exp_sum_s2<<<1, BLK, 0, stream>>>(sumpart, sinv);
    scale_kernel<<<gN, BLK, 0, stream>>>(outp, sinv, N);
}